// QuantizedEmbedding_20555713479188
// MI455X (gfx1250) — compile-verified
//
#include <hip/hip_runtime.h>
#include <hip/hip_bf16.h>
#include <stdint.h>

// QuantizedEmbedding for MI455X (gfx1250, wave32).
//
// Inputs (setup_inputs order):
//   d_in[0] : weight_scale     float32 [1]
//   d_in[1] : quantized_weight int32   [50257, 512]  (each int32 holds a uint8; two 4-bit codes)
//   d_in[2] : input_ids        int32   [8, 2048]     (JAX default x32)
//   d_in[3] : zero_point       int32   [1]
// Output:
//   d_out   : float32 [8, 2048, 1024]
//
// Pure gather + nibble dequant: bandwidth bound (~96 MiB moved, ~4.3 us HBM
// floor; table is ~98 MiB so it largely lives in the 192 MB L2 and the real
// bound is the 64 MiB of output stores). One 256-thread block per token;
// thread t does one 8-byte packed load (global_load_b64) and one 16-byte
// non-temporal store (global_store_b128, TH=NT) so the packed table stays
// resident in L2 while the output streams out.

namespace {

constexpr int kEmbeddingDim = 1024;
constexpr int kPackedDim    = kEmbeddingDim / 2;   // 512 int32 "bytes" per row
constexpr int kThreads      = kPackedDim / 2;      // 256 threads: 2 packed ints each

// Native Clang vector types (required by __builtin_nontemporal_store,
// and guarantee single b64/b128 vmem ops).
typedef float    f32x4 __attribute__((ext_vector_type(4)));
typedef int32_t  i32x2 __attribute__((ext_vector_type(2)));

__global__ __launch_bounds__(kThreads)
void quantized_embedding_kernel(const float* __restrict__ scale_p,
                                const int*   __restrict__ qweight,
                                const int*   __restrict__ ids,
                                const int*   __restrict__ zp_p,
                                float*       __restrict__ out) {
    const int token = blockIdx.x;        // one block per token
    const int t     = threadIdx.x;       // 0..255

    // Uniform scalars -> s_load on gfx1250.
    const int   row   = ids[token];
    const float scale = scale_p[0];
    const float bias  = -scale * (float)zp_p[0];   // out = scale*q + bias

    // Two packed int32 "bytes" per thread: bytes 2t and 2t+1 of the row.
    const i32x2* src = reinterpret_cast<const i32x2*>(
                           qweight + (size_t)row * kPackedDim) + t;
    const i32x2 p = *src;   // global_load_b64, served mostly from L2

    const uint32_t b0 = (uint32_t)p.x;   // uint8 value in low 8 bits
    const uint32_t b1 = (uint32_t)p.y;

    // Byte i -> output columns 2i (low nibble), 2i+1 (high nibble).
    f32x4 v;
    v.x = fmaf(scale, (float)( b0        & 15u), bias);
    v.y = fmaf(scale, (float)((b0 >> 4)  & 15u), bias);
    v.z = fmaf(scale, (float)( b1        & 15u), bias);
    v.w = fmaf(scale, (float)((b1 >> 4)  & 15u), bias);

    // Outputs 4t..4t+3 of this token, streamed non-temporally (keep the
    // packed table hot in L2 across replays; output is write-once).
    f32x4* dst = reinterpret_cast<f32x4*>(
                     out + (size_t)token * kEmbeddingDim) + t;
    __builtin_nontemporal_store(v, dst);
}

} // anonymous namespace

extern "C" void kernel_launch(void* const* d_in, const int* in_sizes, int n_in,
                              void* d_out, int out_size, void* d_ws, size_t ws_size,
                              hipStream_t stream) {
    (void)n_in; (void)out_size; (void)d_ws; (void)ws_size;

    const float* scale   = (const float*)d_in[0];
    const int*   qweight = (const int*)  d_in[1];
    const int*   ids     = (const int*)  d_in[2];
    const int*   zp      = (const int*)  d_in[3];
    float*       out     = (float*)      d_out;

    const int num_tokens = in_sizes[2];  // 8 * 2048 = 16384

    quantized_embedding_kernel<<<num_tokens, kThreads, 0, stream>>>(
        scale, qweight, ids, zp, out);
}